// GeminiACFG_49357764166123
// MI455X (gfx1250) — compile-verified
//
#include <hip/hip_runtime.h>
#include <hip/hip_bf16.h>
#include <math.h>

// ---------------------------------------------------------------------------
// GCN (2x GCNConv + tanh) -> concat -> segment-sum pool -> FC, for MI455X.
// Memory-bound: 2x edge aggregation (~0.9 GB traffic each) dominates at
// 23.3 TB/s (~80us floor). Dense GEMMs use V_WMMA_F32_16X16X4_F32 (full fp32
// fidelity; compute is negligible vs. HBM here).
// ---------------------------------------------------------------------------

typedef __attribute__((ext_vector_type(2))) float v2f;
typedef __attribute__((ext_vector_type(8))) float v8f;

#define F_IN    12
#define HID     64
#define FC_DIM  140
#define N_GRAPHS 256

// ---------------------------------------------------------------------------
// Degree with self-loops: deg[d] += 1 for every edge d=dst[e], plus +1/node.
// ---------------------------------------------------------------------------
__global__ void gcn_degree(const int* __restrict__ eidx, int E, int N,
                           float* __restrict__ deg) {
  int t = blockIdx.x * blockDim.x + threadIdx.x;
  if (t >= E + N) return;
  int d = (t < E) ? eidx[E + t] : (t - E);   // eidx row1 = dst
  atomicAdd(&deg[d], 1.0f);
}

__global__ void gcn_dinv(const float* __restrict__ deg,
                         float* __restrict__ dinv, int N) {
  int t = blockIdx.x * blockDim.x + threadIdx.x;
  if (t >= N) return;
  float d = deg[t];
  dinv[t] = (d > 0.0f) ? rsqrtf(d) : 0.0f;
}

// ---------------------------------------------------------------------------
// C[M x 64] = A[M x K] @ B[K x 64], row-major, fp32 WMMA 16x16x4.
// One 16x16 output tile per wave; 8 waves (256 thr) per block.
// A lane layout (16x4 f32): lanes 0-15 M=0..15 hold K = k0 + {0,1},
//                           lanes 16-31 same M hold K = k0 + {2,3}.
// B lane layout (4x16 f32): VGPR0 = rows {k0, k0+2}, VGPR1 = rows {k0+1, k0+3},
//                           N = lane%16.
// D layout: VGPR r -> row r (lanes 0-15) / row r+8 (lanes 16-31), N = lane%16.
// ---------------------------------------------------------------------------
__global__ __launch_bounds__(256) void gcn_gemm_wmma(
    const float* __restrict__ A, const float* __restrict__ B,
    float* __restrict__ C, int M, int K) {
  const int lane = threadIdx.x & 31;
  const int wave = threadIdx.x >> 5;
  const int tile = blockIdx.x * 8 + wave;
  const int total_tiles = (M >> 4) * (HID >> 4);
  if (tile >= total_tiles) return;          // wave-uniform: EXEC stays all-1s

  const int mt   = tile >> 2;               // HID/16 == 4 column tiles
  const int nt   = tile & 3;
  const int half = lane & 15;
  const int grp  = lane >> 4;
  const int mrow = (mt << 4) + half;        // A row for this lane
  const int col  = (nt << 4) + half;        // B/C column for this lane

  v8f acc = {};
  for (int k0 = 0; k0 < K; k0 += 4) {
    const int kk = k0 + (grp << 1);
    v2f a, b;
    a.x = A[mrow * K + kk];
    a.y = A[mrow * K + kk + 1];
    b.x = B[kk * HID + col];
    b.y = B[(kk + 1) * HID + col];
    acc = __builtin_amdgcn_wmma_f32_16x16x4_f32(
        /*neg_a=*/false, a, /*neg_b=*/false, b,
        /*c_mod=*/(short)0, acc, /*reuse_a=*/false, /*reuse_b=*/false);
  }

  const int rbase = (mt << 4) + (grp << 3);
#pragma unroll
  for (int r = 0; r < 8; ++r)
    C[(rbase + r) * HID + col] = acc[r];
}

// ---------------------------------------------------------------------------
// Edge aggregation: out[dst] += h[src] * dinv[src]*dinv[dst], incl self-loops.
// One thread per (edge, 4-feature chunk): float4 gather + 4 f32 atomics.
// ---------------------------------------------------------------------------
__global__ void gcn_aggregate(const int* __restrict__ eidx, int E, int N,
                              const float* __restrict__ dinv,
                              const float* __restrict__ h,
                              float* __restrict__ out) {
  long long t = (long long)blockIdx.x * blockDim.x + threadIdx.x;
  if (t >= (long long)(E + N) * 16) return;
  const int e = (int)(t >> 4);
  const int c = (int)(t & 15);
  int s, d;
  if (e < E) { s = eidx[e]; d = eidx[E + e]; }
  else       { s = e - E;   d = s; }
  const float norm = dinv[s] * dinv[d];
  const float4 v = reinterpret_cast<const float4*>(h)[s * 16 + c];
  float* o = out + (size_t)d * HID + c * 4;
  atomicAdd(o + 0, v.x * norm);
  atomicAdd(o + 1, v.y * norm);
  atomicAdd(o + 2, v.z * norm);
  atomicAdd(o + 3, v.w * norm);
}

__global__ void gcn_bias_tanh(float* __restrict__ h,
                              const float* __restrict__ b, int n) {
  int t = blockIdx.x * blockDim.x + threadIdx.x;
  if (t >= n) return;
  h[t] = tanhf(h[t] + b[t & (HID - 1)]);
}

// ---------------------------------------------------------------------------
// Segment-sum pool of cat = [x | h1 | h2] over sorted batch ids.
// Block handles 256 consecutive nodes; thread j owns column j; register
// accumulate, atomic flush only on graph-id change.
// ---------------------------------------------------------------------------
__global__ __launch_bounds__(160) void gcn_pool(
    const float* __restrict__ x, const float* __restrict__ h1,
    const float* __restrict__ h2, const int* __restrict__ batch,
    int N, float* __restrict__ pooled) {
  const int j = threadIdx.x;
  const int start = blockIdx.x * 256;
  if (start >= N) return;
  const int end = (start + 256 < N) ? (start + 256) : N;
  const bool act = (j < FC_DIM);

  float acc = 0.0f;
  int cur = batch[start];
  for (int node = start; node < end; ++node) {
    const int g = batch[node];
    if (g != cur) {
      if (act) atomicAdd(&pooled[cur * FC_DIM + j], acc);
      acc = 0.0f;
      cur = g;
    }
    if (act) {
      float v;
      if (j < F_IN)            v = x[node * F_IN + j];
      else if (j < F_IN + HID) v = h1[(size_t)node * HID + (j - F_IN)];
      else                     v = h2[(size_t)node * HID + (j - F_IN - HID)];
      acc += v;
    }
  }
  if (act) atomicAdd(&pooled[cur * FC_DIM + j], acc);
}

// out[g][i] = fc_b[i] + sum_j pooled[g][j] * fc_W[i][j]   (pooled @ fc_W^T + b)
__global__ __launch_bounds__(160) void gcn_fc(
    const float* __restrict__ pooled, const float* __restrict__ W,
    const float* __restrict__ b, float* __restrict__ out) {
  const int i = threadIdx.x;
  const int g = blockIdx.x;
  if (i >= FC_DIM) return;
  const float* p = pooled + g * FC_DIM;
  const float* w = W + i * FC_DIM;
  float s = b[i];
#pragma unroll 4
  for (int j = 0; j < FC_DIM; ++j) s += p[j] * w[j];
  out[g * FC_DIM + i] = s;
}

// ---------------------------------------------------------------------------
extern "C" void kernel_launch(void* const* d_in, const int* in_sizes, int n_in,
                              void* d_out, int out_size, void* d_ws,
                              size_t ws_size, hipStream_t stream) {
  const float* x   = (const float*)d_in[0];
  const float* W1  = (const float*)d_in[1];
  const float* b1  = (const float*)d_in[2];
  const float* W2  = (const float*)d_in[3];
  const float* b2  = (const float*)d_in[4];
  const float* fcW = (const float*)d_in[5];
  const float* fcb = (const float*)d_in[6];
  const int* eidx  = (const int*)d_in[7];   // [2,E] flat: row0=src, row1=dst
  const int* batch = (const int*)d_in[8];
  // d_in[9] edge_index_cg: unused by the reference computation.

  const int N = in_sizes[0] / F_IN;
  const int E = in_sizes[7] / 2;
  float* out = (float*)d_out;

  // Workspace layout (floats): deg | dinv | pre(N*64, reused) | h1 | h2 | pooled
  float* ws     = (float*)d_ws;
  float* deg    = ws;
  float* dinv   = deg + N;
  float* pre    = dinv + N;
  float* h1     = pre + (size_t)N * HID;
  float* h2     = h1 + (size_t)N * HID;
  float* pooled = h2 + (size_t)N * HID;

  hipMemsetAsync(deg,    0, (size_t)N * sizeof(float), stream);
  hipMemsetAsync(h1,     0, (size_t)N * HID * sizeof(float), stream);
  hipMemsetAsync(h2,     0, (size_t)N * HID * sizeof(float), stream);
  hipMemsetAsync(pooled, 0, (size_t)N_GRAPHS * FC_DIM * sizeof(float), stream);

  gcn_degree<<<(E + N + 255) / 256, 256, 0, stream>>>(eidx, E, N, deg);
  gcn_dinv<<<(N + 255) / 256, 256, 0, stream>>>(deg, dinv, N);

  const int tiles   = (N / 16) * (HID / 16);
  const int gblocks = (tiles + 7) / 8;
  const long long aggT = (long long)(E + N) * 16;
  const int ablocks = (int)((aggT + 255) / 256);

  // Layer 1: pre = x @ W1 ; h1 = tanh(aggregate(pre) + b1)
  gcn_gemm_wmma<<<gblocks, 256, 0, stream>>>(x, W1, pre, N, F_IN);
  gcn_aggregate<<<ablocks, 256, 0, stream>>>(eidx, E, N, dinv, pre, h1);
  gcn_bias_tanh<<<(N * HID + 255) / 256, 256, 0, stream>>>(h1, b1, N * HID);

  // Layer 2: pre = h1 @ W2 ; h2 = tanh(aggregate(pre) + b2)
  gcn_gemm_wmma<<<gblocks, 256, 0, stream>>>(h1, W2, pre, N, HID);
  gcn_aggregate<<<ablocks, 256, 0, stream>>>(eidx, E, N, dinv, pre, h2);
  gcn_bias_tanh<<<(N * HID + 255) / 256, 256, 0, stream>>>(h2, b2, N * HID);

  // Pool + FC
  gcn_pool<<<(N + 255) / 256, 160, 0, stream>>>(x, h1, h2, batch, N, pooled);
  gcn_fc<<<N_GRAPHS, 160, 0, stream>>>(pooled, fcW, fcb, out);
}